// LinearRNN_21715354648856
// MI455X (gfx1250) — compile-verified
//
#include <hip/hip_runtime.h>

typedef float v2f __attribute__((ext_vector_type(2)));
typedef float v8f __attribute__((ext_vector_type(8)));
typedef unsigned int u32x4 __attribute__((ext_vector_type(4)));
typedef int i32x4 __attribute__((ext_vector_type(4)));
typedef int i32x8 __attribute__((ext_vector_type(8)));

#define BATCH   32
#define TSTEPS  2048
#define NDIM    256
#define NCHUNK  32
#define CHUNKLEN 64          // TSTEPS / NCHUNK
#define ROWS    (BATCH * TSTEPS)

__device__ __forceinline__ v8f wmma_f32(v2f a, v2f b, v8f c) {
  // D = A(16x4 f32) * B(4x16 f32) + C(16x16 f32)
  return __builtin_amdgcn_wmma_f32_16x16x4_f32(false, a, false, b, (short)0, c,
                                               false, false);
}

// ---- TDM: DMA a dense 256x256 f32 matrix (256 KB) from global into LDS.
// D# built per CDNA5 ISA 8.3/8.4: group0 = {count/type, lds_addr, gaddr},
// group1 = dims/strides/tile, groups 2/3 + extra group unused (zero).
// This toolchain exposes the 6-arg builtin form:
//   (u32x4 g0, i32x8 g1, i32x4 g2, i32x4 g3, i32x8 g4, i32 cpol)
__device__ __forceinline__ void tdm_load_256x256_f32(const float* gsrc,
                                                     float* lds_dst) {
  unsigned long long ga = (unsigned long long)(uintptr_t)gsrc;
  u32x4 g0;
  g0[0] = 1u;                                     // count=1 valid descriptor
  g0[1] = (unsigned int)(uintptr_t)lds_dst;       // LDS byte address
  g0[2] = (unsigned int)(ga & 0xffffffffu);       // global addr [31:0]
  g0[3] = (unsigned int)((ga >> 32) & 0x01ffffffu) | (2u << 30); // [56:32]|type=2
  i32x8 g1;
  g1[0] = (int)(2u << 16);       // workgroup_mask=0, data_size=2 (4 bytes)
  g1[1] = (int)(256u << 16);     // tensor_dim0[15:0]=256 in bits[63:48]
  g1[2] = (int)(256u << 16);     // tensor_dim1[15:0]=256 in bits[95:80]
  g1[3] = (int)(256u << 16);     // tile_dim0=256 in bits[127:112]
  g1[4] = 256;                   // tile_dim1=256, tile_dim2=0
  g1[5] = 256;                   // tensor_dim0_stride = 256 elements
  g1[6] = 0;
  g1[7] = 0;
  i32x4 g2 = {0, 0, 0, 0};
  i32x4 g3 = {0, 0, 0, 0};
  i32x8 g4 = {0, 0, 0, 0, 0, 0, 0, 0};
  __builtin_amdgcn_tensor_load_to_lds(g0, g1, g2, g3, g4, 0);
}

// ---------------------------------------------------------------- transpose
__global__ void k_transpose256(const float* __restrict__ src,
                               float* __restrict__ dst) {
  const int idx = blockIdx.x * 256 + threadIdx.x;   // 65536 threads
  const int i = idx >> 8, j = idx & 255;
  dst[j * NDIM + i] = src[i * NDIM + j];
}

// ------------------------------------------------- Bu = U*Bt, Y0 = U*Dt
// U rows are r = b*T + t.  Bu stored [t][b][n] (time-major, contiguous per
// step).  Register blocked: 2 M-tiles x 2 N-tiles x 2 outputs per wave.
__global__ void k_gemm_u(const float* __restrict__ U,
                         const float* __restrict__ Bt,
                         const float* __restrict__ Dt,
                         float* __restrict__ Bu,
                         float* __restrict__ Y0) {
  const int lane = threadIdx.x & 31, wv = threadIdx.x >> 5;
  const int gw = blockIdx.x * 8 + wv;   // 0..16383
  const int mb = gw >> 3;               // 0..2047 : 32-row block
  const int nb = gw & 7;                // 0..7    : 32-col block
  const int l16 = lane & 15, hi = lane >> 4;
  v8f accB[2][2], accD[2][2];
#pragma unroll
  for (int i = 0; i < 2; ++i)
#pragma unroll
    for (int j = 0; j < 2; ++j) {
      accB[i][j] = (v8f){0,0,0,0,0,0,0,0};
      accD[i][j] = (v8f){0,0,0,0,0,0,0,0};
    }
  const float* urow0 = U + (mb * 32 + l16) * NDIM;
  const float* urow1 = urow0 + 16 * NDIM;
  for (int kk = 0; kk < NDIM; kk += 4) {
    v2f a0 = *(const v2f*)(urow0 + kk + 2 * hi);
    v2f a1 = *(const v2f*)(urow1 + kk + 2 * hi);
#pragma unroll
    for (int j = 0; j < 2; ++j) {
      const int col = nb * 32 + j * 16 + l16;
      v2f bB, bD;
      bB.x = Bt[(kk + 2 * hi) * NDIM + col];
      bB.y = Bt[(kk + 2 * hi + 1) * NDIM + col];
      bD.x = Dt[(kk + 2 * hi) * NDIM + col];
      bD.y = Dt[(kk + 2 * hi + 1) * NDIM + col];
      accB[0][j] = wmma_f32(a0, bB, accB[0][j]);
      accB[1][j] = wmma_f32(a1, bB, accB[1][j]);
      accD[0][j] = wmma_f32(a0, bD, accD[0][j]);
      accD[1][j] = wmma_f32(a1, bD, accD[1][j]);
    }
  }
#pragma unroll
  for (int i = 0; i < 2; ++i)
#pragma unroll
    for (int j = 0; j < 2; ++j) {
      const int col = nb * 32 + j * 16 + l16;
#pragma unroll
      for (int r = 0; r < 8; ++r) {
        const int row = mb * 32 + i * 16 + r + 8 * hi;   // = b*T + t
        const int t = row & (TSTEPS - 1);
        const int b = row >> 11;
        Bu[(t * BATCH + b) * NDIM + col] = accB[i][j][r];
        Y0[row * NDIM + col] = accD[i][j][r];
      }
    }
}

// ---------------------------------------------------------------- Q = P*P
__global__ void k_matsq(const float* __restrict__ P, float* __restrict__ Q) {
  const int lane = threadIdx.x & 31, wv = threadIdx.x >> 5;
  const int gw = blockIdx.x * 8 + wv;             // 0..255
  const int mt = gw >> 4, nt = gw & 15;
  const int l16 = lane & 15, hi = lane >> 4, col0 = nt * 16;
  v8f acc = {0,0,0,0,0,0,0,0};
  const float* prow = P + (mt * 16 + l16) * NDIM;
  for (int kk = 0; kk < NDIM; kk += 4) {
    v2f a = *(const v2f*)(prow + kk + 2 * hi);
    v2f b;
    b.x = P[(kk + 2 * hi) * NDIM + col0 + l16];
    b.y = P[(kk + 2 * hi + 1) * NDIM + col0 + l16];
    acc = wmma_f32(a, b, acc);
  }
#pragma unroll
  for (int r = 0; r < 8; ++r)
    Q[(mt * 16 + r + 8 * hi) * NDIM + col0 + l16] = acc[r];
}

// -------------------------------- per-chunk zero-start scan -> end states E
__global__ void __launch_bounds__(256, 1)
k_chunk_scan_local(const float* __restrict__ Bu, const float* __restrict__ At,
                   float* __restrict__ E) {
  __shared__ float sAt[NDIM * NDIM];     // 256 KB, DMA'd by TDM
  __shared__ float sX[BATCH * NDIM];     // 32 KB
  const int tid = threadIdx.x, g = blockIdx.x;
  if ((tid >> 5) == 0) {                 // wave 0 owns the TDM transfer
    tdm_load_256x256_f32(At, sAt);
    __builtin_amdgcn_s_wait_tensorcnt(0);
  }
  for (int i = tid; i < BATCH * NDIM; i += 256) sX[i] = 0.0f;
  __syncthreads();
  const int lane = tid & 31, wv = tid >> 5;
  const int l16 = lane & 15, hi = lane >> 4;
  const int mt = wv >> 2, ntb = (wv & 3) * 4;
  const int kbeg = g * CHUNKLEN;
  for (int k = kbeg; k < kbeg + CHUNKLEN; ++k) {
    const float* bu = Bu + (k * BATCH) * NDIM;
    v8f acc[4];
#pragma unroll
    for (int i = 0; i < 4; ++i) {
      const int col0 = (ntb + i) * 16;
#pragma unroll
      for (int r = 0; r < 8; ++r)
        acc[i][r] = bu[(mt * 16 + r + 8 * hi) * NDIM + col0 + l16];
    }
    if (k + 1 < kbeg + CHUNKLEN)
      __builtin_prefetch(Bu + ((k + 1) * BATCH + mt * 16) * NDIM, 0, 0);
    const float* xrow = sX + (mt * 16 + l16) * NDIM;
    for (int kk = 0; kk < NDIM; kk += 4) {
      v2f a = *(const v2f*)(xrow + kk + 2 * hi);
#pragma unroll
      for (int i = 0; i < 4; ++i) {
        v2f b;
        b.x = sAt[(kk + 2 * hi) * NDIM + (ntb + i) * 16 + l16];
        b.y = sAt[(kk + 2 * hi + 1) * NDIM + (ntb + i) * 16 + l16];
        acc[i] = wmma_f32(a, b, acc[i]);
      }
    }
    __syncthreads();
#pragma unroll
    for (int i = 0; i < 4; ++i) {
      const int col0 = (ntb + i) * 16;
#pragma unroll
      for (int r = 0; r < 8; ++r)
        sX[(mt * 16 + r + 8 * hi) * NDIM + col0 + l16] = acc[i][r];
    }
    __syncthreads();
  }
  for (int i = tid; i < BATCH * NDIM; i += 256)
    E[g * BATCH * NDIM + i] = sX[i];
}

// -------------------- serial prefix over chunks: S[g]=x; x = x*AtL + E[g]
__global__ void __launch_bounds__(256, 1)
k_chunk_prefix(const float* __restrict__ AtL, const float* __restrict__ Eg,
               const float* __restrict__ x0, float* __restrict__ S) {
  __shared__ float sA[NDIM * NDIM];
  __shared__ float sX[BATCH * NDIM];
  const int tid = threadIdx.x;
  if ((tid >> 5) == 0) {
    tdm_load_256x256_f32(AtL, sA);
    __builtin_amdgcn_s_wait_tensorcnt(0);
  }
  for (int i = tid; i < BATCH * NDIM; i += 256) sX[i] = x0[i];
  __syncthreads();
  const int lane = tid & 31, wv = tid >> 5;
  const int l16 = lane & 15, hi = lane >> 4;
  const int mt = wv >> 2, ntb = (wv & 3) * 4;
  for (int g = 0; g < NCHUNK; ++g) {
    for (int i = tid; i < BATCH * NDIM; i += 256)
      S[g * BATCH * NDIM + i] = sX[i];
    const float* eg = Eg + g * BATCH * NDIM;
    v8f acc[4];
#pragma unroll
    for (int i = 0; i < 4; ++i) {
      const int col0 = (ntb + i) * 16;
#pragma unroll
      for (int r = 0; r < 8; ++r)
        acc[i][r] = eg[(mt * 16 + r + 8 * hi) * NDIM + col0 + l16];
    }
    const float* xrow = sX + (mt * 16 + l16) * NDIM;
    for (int kk = 0; kk < NDIM; kk += 4) {
      v2f a = *(const v2f*)(xrow + kk + 2 * hi);
#pragma unroll
      for (int i = 0; i < 4; ++i) {
        v2f b;
        b.x = sA[(kk + 2 * hi) * NDIM + (ntb + i) * 16 + l16];
        b.y = sA[(kk + 2 * hi + 1) * NDIM + (ntb + i) * 16 + l16];
        acc[i] = wmma_f32(a, b, acc[i]);
      }
    }
    __syncthreads();
#pragma unroll
    for (int i = 0; i < 4; ++i) {
      const int col0 = (ntb + i) * 16;
#pragma unroll
      for (int r = 0; r < 8; ++r)
        sX[(mt * 16 + r + 8 * hi) * NDIM + col0 + l16] = acc[i][r];
    }
    __syncthreads();
  }
}

// ------------- chunk rerun from true start state; overwrite Bu[k] with x_k
__global__ void __launch_bounds__(256, 1)
k_chunk_scan_out(float* __restrict__ BuX, const float* __restrict__ At,
                 const float* __restrict__ S) {
  __shared__ float sAt[NDIM * NDIM];
  __shared__ float sX[BATCH * NDIM];
  const int tid = threadIdx.x, g = blockIdx.x;
  if ((tid >> 5) == 0) {
    tdm_load_256x256_f32(At, sAt);
    __builtin_amdgcn_s_wait_tensorcnt(0);
  }
  for (int i = tid; i < BATCH * NDIM; i += 256)
    sX[i] = S[g * BATCH * NDIM + i];
  __syncthreads();
  const int lane = tid & 31, wv = tid >> 5;
  const int l16 = lane & 15, hi = lane >> 4;
  const int mt = wv >> 2, ntb = (wv & 3) * 4;
  const int kbeg = g * CHUNKLEN;
  for (int k = kbeg; k < kbeg + CHUNKLEN; ++k) {
    float* bu = BuX + (k * BATCH) * NDIM;
    v8f acc[4];
#pragma unroll
    for (int i = 0; i < 4; ++i) {
      const int col0 = (ntb + i) * 16;
#pragma unroll
      for (int r = 0; r < 8; ++r)
        acc[i][r] = bu[(mt * 16 + r + 8 * hi) * NDIM + col0 + l16];
    }
    // make sure Bu[k] loads landed before we overwrite the same addresses
    asm volatile("s_wait_loadcnt 0" ::: "memory");
    // write x_k (pre-update state) over Bu[k]  -> becomes Xtraj[t][b][n]
#pragma unroll
    for (int i = 0; i < 4; ++i) {
      const int col0 = (ntb + i) * 16;
#pragma unroll
      for (int r = 0; r < 8; ++r) {
        const int b = mt * 16 + r + 8 * hi;
        bu[b * NDIM + col0 + l16] = sX[b * NDIM + col0 + l16];
      }
    }
    if (k + 1 < kbeg + CHUNKLEN)
      __builtin_prefetch(BuX + ((k + 1) * BATCH + mt * 16) * NDIM, 0, 0);
    const float* xrow = sX + (mt * 16 + l16) * NDIM;
    for (int kk = 0; kk < NDIM; kk += 4) {
      v2f a = *(const v2f*)(xrow + kk + 2 * hi);
#pragma unroll
      for (int i = 0; i < 4; ++i) {
        v2f b;
        b.x = sAt[(kk + 2 * hi) * NDIM + (ntb + i) * 16 + l16];
        b.y = sAt[(kk + 2 * hi + 1) * NDIM + (ntb + i) * 16 + l16];
        acc[i] = wmma_f32(a, b, acc[i]);
      }
    }
    __syncthreads();
#pragma unroll
    for (int i = 0; i < 4; ++i) {
      const int col0 = (ntb + i) * 16;
#pragma unroll
      for (int r = 0; r < 8; ++r)
        sX[(mt * 16 + r + 8 * hi) * NDIM + col0 + l16] = acc[i][r];
    }
    __syncthreads();
  }
}

// --------------------------- Y[b*T+t] += Xtraj[t*32+b] * Ct  (Y holds U*Dt)
// Register blocked: 2 M-tiles x 2 N-tiles per wave.
__global__ void k_gemm_y(const float* __restrict__ Xtraj,
                         const float* __restrict__ Ct,
                         float* __restrict__ Y) {
  const int lane = threadIdx.x & 31, wv = threadIdx.x >> 5;
  const int gw = blockIdx.x * 8 + wv;   // 0..16383
  const int mb = gw >> 3;               // 0..2047
  const int nb = gw & 7;                // 0..7
  const int l16 = lane & 15, hi = lane >> 4;
  v8f acc[2][2];
#pragma unroll
  for (int i = 0; i < 2; ++i)
#pragma unroll
    for (int j = 0; j < 2; ++j) {
      const int col = nb * 32 + j * 16 + l16;
#pragma unroll
      for (int r = 0; r < 8; ++r) {
        const int rp = mb * 32 + i * 16 + r + 8 * hi;  // Xtraj row = t*32 + b
        const int t = rp >> 5, b = rp & 31;
        acc[i][j][r] = Y[(b * TSTEPS + t) * NDIM + col];
      }
    }
  const float* xrow0 = Xtraj + (mb * 32 + l16) * NDIM;
  const float* xrow1 = xrow0 + 16 * NDIM;
  for (int kk = 0; kk < NDIM; kk += 4) {
    v2f a0 = *(const v2f*)(xrow0 + kk + 2 * hi);
    v2f a1 = *(const v2f*)(xrow1 + kk + 2 * hi);
#pragma unroll
    for (int j = 0; j < 2; ++j) {
      const int col = nb * 32 + j * 16 + l16;
      v2f b;
      b.x = Ct[(kk + 2 * hi) * NDIM + col];
      b.y = Ct[(kk + 2 * hi + 1) * NDIM + col];
      acc[0][j] = wmma_f32(a0, b, acc[0][j]);
      acc[1][j] = wmma_f32(a1, b, acc[1][j]);
    }
  }
#pragma unroll
  for (int i = 0; i < 2; ++i)
#pragma unroll
    for (int j = 0; j < 2; ++j) {
      const int col = nb * 32 + j * 16 + l16;
#pragma unroll
      for (int r = 0; r < 8; ++r) {
        const int rp = mb * 32 + i * 16 + r + 8 * hi;
        const int t = rp >> 5, b = rp & 31;
        Y[(b * TSTEPS + t) * NDIM + col] = acc[i][j][r];
      }
    }
}

// ---------------------------------------------------------------- launcher
extern "C" void kernel_launch(void* const* d_in, const int* in_sizes, int n_in,
                              void* d_out, int out_size, void* d_ws,
                              size_t ws_size, hipStream_t stream) {
  (void)in_sizes; (void)n_in; (void)out_size; (void)ws_size;
  const float* U  = (const float*)d_in[0];
  const float* x0 = (const float*)d_in[1];
  const float* A  = (const float*)d_in[2];
  const float* Bm = (const float*)d_in[3];
  const float* Cm = (const float*)d_in[4];
  const float* Dm = (const float*)d_in[5];
  float* Y = (float*)d_out;

  float* W   = (float*)d_ws;
  float* Bu  = W;                        // 16,777,216 f (aliased as Xtraj later)
  float* At  = W + 16777216;             // 65536 f each
  float* Btr = At + 65536;
  float* Ctr = Btr + 65536;
  float* Dtr = Ctr + 65536;
  float* E   = Dtr + 65536;              // 262,144 f
  float* S   = E + NCHUNK * BATCH * NDIM;
  float* P0  = S + NCHUNK * BATCH * NDIM;
  float* P1  = P0 + 65536;

  k_transpose256<<<256, 256, 0, stream>>>(A, At);
  k_transpose256<<<256, 256, 0, stream>>>(Bm, Btr);
  k_transpose256<<<256, 256, 0, stream>>>(Cm, Ctr);
  k_transpose256<<<256, 256, 0, stream>>>(Dm, Dtr);

  k_gemm_u<<<2048, 256, 0, stream>>>(U, Btr, Dtr, Bu, Y);

  // (A^T)^64 by repeated squaring: At -> P1 -> P0 -> ... -> P0
  k_matsq<<<32, 256, 0, stream>>>(At, P1);
  k_matsq<<<32, 256, 0, stream>>>(P1, P0);
  k_matsq<<<32, 256, 0, stream>>>(P0, P1);
  k_matsq<<<32, 256, 0, stream>>>(P1, P0);
  k_matsq<<<32, 256, 0, stream>>>(P0, P1);
  k_matsq<<<32, 256, 0, stream>>>(P1, P0);

  k_chunk_scan_local<<<NCHUNK, 256, 0, stream>>>(Bu, At, E);
  k_chunk_prefix<<<1, 256, 0, stream>>>(P0, E, x0, S);
  k_chunk_scan_out<<<NCHUNK, 256, 0, stream>>>(Bu, At, S);
  k_gemm_y<<<2048, 256, 0, stream>>>(Bu, Ctr, Y);
}